// AFNO2D_62234076119789
// MI455X (gfx1250) — compile-verified
//
#include <hip/hip_runtime.h>

// AFNO2D on gfx1250.
// Pipeline: rowFFT(W) -> colFFT(H) -> WMMA block-diag complex MLP (in place)
//           -> colIFFT(H) -> rowIFFT(W, real out).
// d_ws layout: Re plane (B*C*H*W floats) then Im plane (256 MB each).

#define B_   4
#define HH   256
#define WW   256
#define CC   256
#define NB   8
#define BS   32
#define HW_  65536
#define PI_F 3.14159265358979323846f

typedef __attribute__((ext_vector_type(2))) float v2f;
typedef __attribute__((ext_vector_type(8))) float v8f;

__device__ __forceinline__ float sshrink(float v) {
  float t = fabsf(v) - 0.01f;
  return t > 0.f ? copysignf(t, v) : 0.f;
}

// ---------------------------------------------------------------------------
// Forward FFT along W. Input x is (B, H, W, C) real; output planar (B,C,H,W).
// One workgroup: one (b,h) line-group, 8 channels (2 lines per pass).
// 256-pt radix-2 Stockham, LDS ping-pong, twiddles from LDS table.
// ---------------------------------------------------------------------------
__global__ __launch_bounds__(256) void afno_fft_rows_fwd(
    const float* __restrict__ x, float* __restrict__ Re, float* __restrict__ Im) {
  __shared__ float sR[2][2][WW];
  __shared__ float sI[2][2][WW];
  __shared__ float twc[128], tws[128];
  const int bh = blockIdx.x;                 // b*H + h
  const int c0base = blockIdx.y * 8;
  const int tid = threadIdx.x;
  const int w = tid;
  if (tid < 128) {                           // e^{-2*pi*i*t/256}
    float c, s;
    __sincosf(-(2.f * PI_F / 256.f) * (float)tid, &s, &c);
    twc[tid] = c; tws[tid] = s;
  }
  __syncthreads();
  for (int pc = 0; pc < 4; ++pc) {
    const int c0 = c0base + pc * 2;
    #pragma unroll
    for (int cc = 0; cc < 2; ++cc) {
      sR[0][cc][w] = x[((size_t)bh * WW + w) * CC + (c0 + cc)];
      sI[0][cc][w] = 0.f;
    }
    __syncthreads();
    int cur = 0;
    #pragma unroll
    for (int t = 0; t < 8; ++t) {
      const int line = tid >> 7;
      const int j = tid & 127;
      const int m = 1 << t;
      const int l = 128 >> t;
      const int q = j & (m - 1);
      const int p = j >> t;
      const float ar = sR[cur][line][q + m * p];
      const float ai = sI[cur][line][q + m * p];
      const float br = sR[cur][line][q + m * (p + l)];
      const float bi = sI[cur][line][q + m * (p + l)];
      const float cw = twc[p << t], swv = tws[p << t];   // angle = -pi*p/l
      const float dr = ar - br, di = ai - bi;
      sR[cur ^ 1][line][q + 2 * m * p] = ar + br;
      sI[cur ^ 1][line][q + 2 * m * p] = ai + bi;
      sR[cur ^ 1][line][q + m * (2 * p + 1)] = dr * cw - di * swv;
      sI[cur ^ 1][line][q + m * (2 * p + 1)] = dr * swv + di * cw;
      cur ^= 1;
      __syncthreads();
    }
    #pragma unroll
    for (int cc = 0; cc < 2; ++cc) {
      const size_t base = ((size_t)((bh >> 8) * CC + c0 + cc)) * HW_
                        + (size_t)(bh & 255) * WW;
      Re[base + w] = sR[0][cc][w] * 0.0625f;   // ortho: 1/sqrt(256) per axis
      Im[base + w] = sI[0][cc][w] * 0.0625f;
    }
    __syncthreads();
  }
}

// ---------------------------------------------------------------------------
// Complex FFT along H, in place on planar Re/Im. sgn=-1 fwd, +1 inverse.
// Tile: 8 w-columns x 256 rows per workgroup; all columns per stage.
// ---------------------------------------------------------------------------
__global__ __launch_bounds__(256) void afno_fft_cols(
    float* __restrict__ Re, float* __restrict__ Im, float sgn) {
  __shared__ float tR[2][HH][8];
  __shared__ float tI[2][HH][8];
  __shared__ float twc[128], tws[128];
  const int bc = blockIdx.x;                 // b*C + c
  const int w0 = blockIdx.y * 8;
  const size_t base = (size_t)bc * HW_ + w0;
  const int tid = threadIdx.x;
  const int lw = tid & 7;
  const int lr = tid >> 3;                   // 0..31
  if (tid < 128) {
    float c, s;
    __sincosf(sgn * (2.f * PI_F / 256.f) * (float)tid, &s, &c);
    twc[tid] = c; tws[tid] = s;
  }
  #pragma unroll
  for (int r = 0; r < 8; ++r) {
    const int h = r * 32 + lr;
    tR[0][h][lw] = Re[base + (size_t)h * WW + lw];
    tI[0][h][lw] = Im[base + (size_t)h * WW + lw];
  }
  __syncthreads();
  int cur = 0;
  #pragma unroll
  for (int t = 0; t < 8; ++t) {
    const int m = 1 << t;
    const int l = 128 >> t;
    #pragma unroll
    for (int r = 0; r < 4; ++r) {            // 1024 butterflies / 256 threads
      const int idx = r * 256 + tid;
      const int col = idx & 7;
      const int j = idx >> 3;
      const int q = j & (m - 1);
      const int p = j >> t;
      const float ar = tR[cur][q + m * p][col];
      const float ai = tI[cur][q + m * p][col];
      const float br = tR[cur][q + m * (p + l)][col];
      const float bi = tI[cur][q + m * (p + l)][col];
      const float cw = twc[p << t], swv = tws[p << t];
      const float dr = ar - br, di = ai - bi;
      tR[cur ^ 1][q + 2 * m * p][col] = ar + br;
      tI[cur ^ 1][q + 2 * m * p][col] = ai + bi;
      tR[cur ^ 1][q + m * (2 * p + 1)][col] = dr * cw - di * swv;
      tI[cur ^ 1][q + m * (2 * p + 1)][col] = dr * swv + di * cw;
    }
    cur ^= 1;
    __syncthreads();
  }
  #pragma unroll
  for (int r = 0; r < 8; ++r) {
    const int h = r * 32 + lr;
    Re[base + (size_t)h * WW + lw] = tR[0][h][lw] * 0.0625f;
    Im[base + (size_t)h * WW + lw] = tI[0][h][lw] * 0.0625f;
  }
}

// ---------------------------------------------------------------------------
// Block-diagonal complex MLP at every frequency, in place, fp32 WMMA.
// grid = (pointTiles, block k). Wave unit: M=16 points x one 32-ch block.
// A layout (f32 16x16x4): lanes0-15 M=0..15 hold K={k0,k0+1}; lanes16-31
// hold K={k0+2,k0+3}. B mirrored with N across lanes. D: VGPR g -> M=g(+8),
// N = lane&15 (so per-lane bias splat into C is exact).
// ---------------------------------------------------------------------------
__global__ __launch_bounds__(256) void afno_mix(
    float* __restrict__ Re, float* __restrict__ Im,
    const float* __restrict__ w1, const float* __restrict__ b1,
    const float* __restrict__ w2, const float* __restrict__ b2) {
  __shared__ float sw[4][BS * BS];      // w1r,w1i,w2r,w2i for block k (16 KB)
  __shared__ float sb[4][BS];
  __shared__ float stg[8][2][BS * 16];  // per-wave [r/i][ch][pt] (32 KB)
  const int k = blockIdx.y;
  const int tid = threadIdx.x;
  for (int i = tid; i < BS * BS; i += 256) {
    sw[0][i] = w1[(size_t)k * 1024 + i];
    sw[1][i] = w1[(size_t)(NB + k) * 1024 + i];
    sw[2][i] = w2[(size_t)k * 1024 + i];
    sw[3][i] = w2[(size_t)(NB + k) * 1024 + i];
  }
  if (tid < BS) {
    sb[0][tid] = b1[k * BS + tid];
    sb[1][tid] = b1[(NB + k) * BS + tid];
    sb[2][tid] = b2[k * BS + tid];
    sb[3][tid] = b2[(NB + k) * BS + tid];
  }
  __syncthreads();

  const int wave = tid >> 5, lane = tid & 31;
  const int half = lane >> 4, m = lane & 15;
  const size_t pt = (size_t)blockIdx.x * 128 + (size_t)wave * 16;  // flat b*HW+hw
  const size_t planeBase =
      ((size_t)((pt >> 16) * CC + k * BS)) * HW_ + (pt & 65535);
  float* const str = &stg[wave][0][0];
  float* const sti = &stg[wave][1][0];

  // A matrices in register, WMMA A layout: aX[2s],aX[2s+1] = A[m][4s+2*half +0,1]
  float aR[16], aI[16];
  #pragma unroll
  for (int s = 0; s < 8; ++s) {
    const int ch = 4 * s + 2 * half;
    aR[2*s]   = Re[planeBase + (size_t)ch * HW_ + m];
    aR[2*s+1] = Re[planeBase + (size_t)(ch + 1) * HW_ + m];
    aI[2*s]   = Im[planeBase + (size_t)ch * HW_ + m];
    aI[2*s+1] = Im[planeBase + (size_t)(ch + 1) * HW_ + m];
  }

  // ---- layer 1: o1r = relu(re*W1r - im*W1i + b1r), o1i = relu(im*W1r + re*W1i + b1i)
  #pragma unroll
  for (int nh = 0; nh < 2; ++nh) {
    const int n = nh * 16 + m;                       // output channel in block
    const float bR = sb[0][n], bI = sb[1][n];
    v8f accA = {bR, bR, bR, bR, bR, bR, bR, bR};
    v8f accB = {};
    v8f accC = {bI, bI, bI, bI, bI, bI, bI, bI};
    v8f accD = {};
    #pragma unroll
    for (int s = 0; s < 8; ++s) {
      const int kk = 4 * s + 2 * half;
      v2f ar, ai, wr, wi;
      ar[0] = aR[2*s]; ar[1] = aR[2*s+1];
      ai[0] = aI[2*s]; ai[1] = aI[2*s+1];
      wr[0] = sw[0][kk * BS + n]; wr[1] = sw[0][(kk + 1) * BS + n];
      wi[0] = sw[1][kk * BS + n]; wi[1] = sw[1][(kk + 1) * BS + n];
      accA = __builtin_amdgcn_wmma_f32_16x16x4_f32(false, ar, false, wr, (short)0, accA, false, false);
      accB = __builtin_amdgcn_wmma_f32_16x16x4_f32(false, ai, false, wi, (short)0, accB, false, false);
      accC = __builtin_amdgcn_wmma_f32_16x16x4_f32(false, ai, false, wr, (short)0, accC, false, false);
      accD = __builtin_amdgcn_wmma_f32_16x16x4_f32(false, ar, false, wi, (short)0, accD, false, false);
    }
    #pragma unroll
    for (int g = 0; g < 8; ++g) {                    // D: (M=g+8*half, N=n)
      const int M = g + 8 * half;
      str[n * 16 + M] = fmaxf(accA[g] - accB[g], 0.f);
      sti[n * 16 + M] = fmaxf(accC[g] + accD[g], 0.f);
    }
  }
  __syncthreads();

  // ---- transpose D layout -> A layout through LDS
  float cR[16], cI[16];
  #pragma unroll
  for (int s = 0; s < 8; ++s) {
    const int kk = 4 * s + 2 * half;
    cR[2*s]   = str[kk * 16 + m];
    cR[2*s+1] = str[(kk + 1) * 16 + m];
    cI[2*s]   = sti[kk * 16 + m];
    cI[2*s+1] = sti[(kk + 1) * 16 + m];
  }

  // ---- layer 2 + softshrink, result back into stage in [ch][pt] layout
  #pragma unroll
  for (int nh = 0; nh < 2; ++nh) {
    const int n = nh * 16 + m;
    const float bR = sb[2][n], bI = sb[3][n];
    v8f accA = {bR, bR, bR, bR, bR, bR, bR, bR};
    v8f accB = {};
    v8f accC = {bI, bI, bI, bI, bI, bI, bI, bI};
    v8f accD = {};
    #pragma unroll
    for (int s = 0; s < 8; ++s) {
      const int kk = 4 * s + 2 * half;
      v2f ar, ai, wr, wi;
      ar[0] = cR[2*s]; ar[1] = cR[2*s+1];
      ai[0] = cI[2*s]; ai[1] = cI[2*s+1];
      wr[0] = sw[2][kk * BS + n]; wr[1] = sw[2][(kk + 1) * BS + n];
      wi[0] = sw[3][kk * BS + n]; wi[1] = sw[3][(kk + 1) * BS + n];
      accA = __builtin_amdgcn_wmma_f32_16x16x4_f32(false, ar, false, wr, (short)0, accA, false, false);
      accB = __builtin_amdgcn_wmma_f32_16x16x4_f32(false, ai, false, wi, (short)0, accB, false, false);
      accC = __builtin_amdgcn_wmma_f32_16x16x4_f32(false, ai, false, wr, (short)0, accC, false, false);
      accD = __builtin_amdgcn_wmma_f32_16x16x4_f32(false, ar, false, wi, (short)0, accD, false, false);
    }
    #pragma unroll
    for (int g = 0; g < 8; ++g) {
      const int M = g + 8 * half;
      str[n * 16 + M] = sshrink(accA[g] - accB[g]);
      sti[n * 16 + M] = sshrink(accC[g] + accD[g]);
    }
  }
  __syncthreads();

  // ---- in-place store, transposed so each lane writes 4x float4 per plane
  #pragma unroll
  for (int q4 = 0; q4 < 4; ++q4) {
    float4 vr, vi;
    vr.x = str[lane * 16 + q4 * 4 + 0]; vr.y = str[lane * 16 + q4 * 4 + 1];
    vr.z = str[lane * 16 + q4 * 4 + 2]; vr.w = str[lane * 16 + q4 * 4 + 3];
    vi.x = sti[lane * 16 + q4 * 4 + 0]; vi.y = sti[lane * 16 + q4 * 4 + 1];
    vi.z = sti[lane * 16 + q4 * 4 + 2]; vi.w = sti[lane * 16 + q4 * 4 + 3];
    *reinterpret_cast<float4*>(&Re[planeBase + (size_t)lane * HW_ + q4 * 4]) = vr;
    *reinterpret_cast<float4*>(&Im[planeBase + (size_t)lane * HW_ + q4 * 4]) = vi;
  }
}

// ---------------------------------------------------------------------------
// Inverse FFT along W; write real part to out (B, H*W, C).
// ---------------------------------------------------------------------------
__global__ __launch_bounds__(256) void afno_ifft_rows(
    const float* __restrict__ Re, const float* __restrict__ Im,
    float* __restrict__ out) {
  __shared__ float sR[2][2][WW];
  __shared__ float sI[2][2][WW];
  __shared__ float twc[128], tws[128];
  const int bh = blockIdx.x;
  const int c0base = blockIdx.y * 8;
  const int tid = threadIdx.x;
  const int w = tid;
  if (tid < 128) {
    float c, s;
    __sincosf((2.f * PI_F / 256.f) * (float)tid, &s, &c);
    twc[tid] = c; tws[tid] = s;
  }
  __syncthreads();
  for (int pc = 0; pc < 4; ++pc) {
    const int c0 = c0base + pc * 2;
    #pragma unroll
    for (int cc = 0; cc < 2; ++cc) {
      const size_t base = ((size_t)((bh >> 8) * CC + c0 + cc)) * HW_
                        + (size_t)(bh & 255) * WW;
      sR[0][cc][w] = Re[base + w];
      sI[0][cc][w] = Im[base + w];
    }
    __syncthreads();
    int cur = 0;
    #pragma unroll
    for (int t = 0; t < 8; ++t) {
      const int line = tid >> 7;
      const int j = tid & 127;
      const int m = 1 << t;
      const int l = 128 >> t;
      const int q = j & (m - 1);
      const int p = j >> t;
      const float ar = sR[cur][line][q + m * p];
      const float ai = sI[cur][line][q + m * p];
      const float br = sR[cur][line][q + m * (p + l)];
      const float bi = sI[cur][line][q + m * (p + l)];
      const float cw = twc[p << t], swv = tws[p << t];
      const float dr = ar - br, di = ai - bi;
      sR[cur ^ 1][line][q + 2 * m * p] = ar + br;
      sI[cur ^ 1][line][q + 2 * m * p] = ai + bi;
      sR[cur ^ 1][line][q + m * (2 * p + 1)] = dr * cw - di * swv;
      sI[cur ^ 1][line][q + m * (2 * p + 1)] = dr * swv + di * cw;
      cur ^= 1;
      __syncthreads();
    }
    #pragma unroll
    for (int cc = 0; cc < 2; ++cc) {
      out[((size_t)bh * WW + w) * CC + (c0 + cc)] = sR[0][cc][w] * 0.0625f;
    }
    __syncthreads();
  }
}

extern "C" void kernel_launch(void* const* d_in, const int* in_sizes, int n_in,
                              void* d_out, int out_size, void* d_ws, size_t ws_size,
                              hipStream_t stream) {
  (void)in_sizes; (void)n_in; (void)out_size; (void)ws_size;
  const float* x  = (const float*)d_in[0];
  const float* w1 = (const float*)d_in[1];
  const float* b1 = (const float*)d_in[2];
  const float* w2 = (const float*)d_in[3];
  const float* b2 = (const float*)d_in[4];
  float* out = (float*)d_out;
  float* Re = (float*)d_ws;                         // needs 2*256MB scratch
  float* Im = Re + (size_t)B_ * CC * HW_;

  afno_fft_rows_fwd<<<dim3(B_ * HH, CC / 8), 256, 0, stream>>>(x, Re, Im);
  afno_fft_cols   <<<dim3(B_ * CC, WW / 8), 256, 0, stream>>>(Re, Im, -1.f);
  afno_mix        <<<dim3((B_ * HW_) / 128, NB), 256, 0, stream>>>(Re, Im, w1, b1, w2, b2);
  afno_fft_cols   <<<dim3(B_ * CC, WW / 8), 256, 0, stream>>>(Re, Im, 1.f);
  afno_ifft_rows  <<<dim3(B_ * HH, CC / 8), 256, 0, stream>>>(Re, Im, out);
}